// edgeNet_27590869910157
// MI455X (gfx1250) — compile-verified
//
#include <hip/hip_runtime.h>

typedef __attribute__((ext_vector_type(2))) float v2f;
typedef __attribute__((ext_vector_type(8))) float v8f;

#define N_NODES 50000
#define D_FEAT  128
#define H1DIM   128
#define H2DIM   64
#define WAVES   4   // waves per block in the node kernel; 1 node-tile (16 nodes) per wave

__device__ __forceinline__ v8f wmma_f32(v2f a, v2f b, v8f c) {
  // D(16x16,f32) = A(16x4,f32) * B(4x16,f32) + C
  return __builtin_amdgcn_wmma_f32_16x16x4_f32(
      /*neg_a=*/false, a, /*neg_b=*/false, b,
      /*c_mod=*/(short)0, c, /*reuse_a=*/false, /*reuse_b=*/false);
}

// One wave computes, for a 16-node tile:
//   H = relu(X @ W + b)          (16x128, staged in LDS)
//   A = H @ Watt (+ att bias)    (16x64, written to workspace)
// done twice (neighbor half -> A1, self half -> A2).
__global__ __launch_bounds__(WAVES * 32)
void node_transform_kernel(const float* __restrict__ features,
                           const float* __restrict__ W_nb,   const float* __restrict__ b_nb,
                           const float* __restrict__ W_self, const float* __restrict__ b_self,
                           const float* __restrict__ W_att1, const float* __restrict__ b_att1,
                           float* __restrict__ A1, float* __restrict__ A2) {
  __shared__ float ldsAll[WAVES * 16 * D_FEAT];   // 8KB per wave, 32KB per block
  const int lane = threadIdx.x & 31;
  const int wave = threadIdx.x >> 5;
  const int tile = blockIdx.x * WAVES + wave;     // wave-uniform -> full EXEC inside
  const int nTiles = N_NODES / 16;                // 3125, exact
  if (tile >= nTiles) return;

  float* ldsH = ldsAll + wave * 16 * D_FEAT;

  const int row  = lane & 15;          // M row (A/C) and N column (B/C) index
  const int kofs = (lane >> 4) << 1;   // lanes 16-31 hold K+2,K+3
  const int hi   = (lane >> 4) << 3;   // C rows: M = r (+8 for upper half-wave)
  const int n0   = tile * 16;

  // Preload X tile as 32 ready-made A-operands (packed 8B loads, K-contiguous).
  v2f xa[32];
  const float* xbase = features + (size_t)(n0 + row) * D_FEAT + kofs;
#pragma unroll
  for (int ks = 0; ks < 32; ++ks)
    xa[ks] = *(const v2f*)(xbase + ks * 4);

  for (int half = 0; half < 2; ++half) {
    const float* W   = half ? W_self : W_nb;
    const float* bb  = half ? b_self : b_nb;
    const float* Wa  = W_att1 + (half ? (size_t)H1DIM * H2DIM : 0); // top/bottom 128 rows
    float*       out = (half ? A2 : A1) + (size_t)n0 * H2DIM;

    // ---- layer 1: H = relu(X @ W + b) -> LDS (8 N-tiles of 16) ----
    for (int j = 0; j < 8; ++j) {
      v8f acc = {};
#pragma unroll
      for (int ks = 0; ks < 32; ++ks) {
        const int k = ks * 4 + kofs;
        v2f bop;
        bop.x = W[(k + 0) * H1DIM + j * 16 + row];
        bop.y = W[(k + 1) * H1DIM + j * 16 + row];
        acc = wmma_f32(xa[ks], bop, acc);
      }
      const float bv = bb[j * 16 + row];
#pragma unroll
      for (int r = 0; r < 8; ++r) {
        float v = acc[r] + bv;
        ldsH[(r + hi) * D_FEAT + j * 16 + row] = v > 0.f ? v : 0.f;
      }
    }

    // ---- layer 2: A = H @ Wa (+ b_att1 folded into A1 only) ----
    for (int j = 0; j < 4; ++j) {
      v8f acc = {};
#pragma unroll
      for (int ks = 0; ks < 32; ++ks) {
        const int k = ks * 4 + kofs;
        v2f aop = *(const v2f*)(ldsH + row * D_FEAT + k);   // 8B LDS load
        v2f bop;
        bop.x = Wa[(k + 0) * H2DIM + j * 16 + row];
        bop.y = Wa[(k + 1) * H2DIM + j * 16 + row];
        acc = wmma_f32(aop, bop, acc);
      }
      const float ab = half ? 0.f : b_att1[j * 16 + row];
#pragma unroll
      for (int r = 0; r < 8; ++r)
        out[(size_t)(r + hi) * H2DIM + j * 16 + row] = acc[r] + ab;
    }
  }
}

// Per edge: w = relu(A1[src] + A2[dst]) . W_att2 + b_att2   (A1 already holds b_att1)
__global__ __launch_bounds__(256)
void edge_weight_kernel(const int* __restrict__ edges,
                        const float* __restrict__ A1, const float* __restrict__ A2,
                        const float* __restrict__ W_att2, const float* __restrict__ b_att2,
                        float* __restrict__ out, int nEdges) {
  int e = blockIdx.x * blockDim.x + threadIdx.x;
  if (e >= nEdges) return;
  const int src = edges[2 * e + 0];
  const int dst = edges[2 * e + 1];
  const float4* p1 = (const float4*)(A1 + (size_t)src * H2DIM);  // 256B rows, aligned
  const float4* p2 = (const float4*)(A2 + (size_t)dst * H2DIM);
  const float4* w  = (const float4*)W_att2;
  float acc = b_att2[0];
#pragma unroll
  for (int i = 0; i < 16; ++i) {
    float4 a = p1[i], b = p2[i], ww = w[i];
    acc += fmaxf(a.x + b.x, 0.f) * ww.x;
    acc += fmaxf(a.y + b.y, 0.f) * ww.y;
    acc += fmaxf(a.z + b.z, 0.f) * ww.z;
    acc += fmaxf(a.w + b.w, 0.f) * ww.w;
  }
  out[e] = acc;
}

extern "C" void kernel_launch(void* const* d_in, const int* in_sizes, int n_in,
                              void* d_out, int out_size, void* d_ws, size_t ws_size,
                              hipStream_t stream) {
  const float* features = (const float*)d_in[0];
  const int*   edges    = (const int*)  d_in[1];
  const float* W_nb     = (const float*)d_in[2];
  const float* b_nb     = (const float*)d_in[3];
  const float* W_self   = (const float*)d_in[4];
  const float* b_self   = (const float*)d_in[5];
  const float* W_att1   = (const float*)d_in[6];
  const float* b_att1   = (const float*)d_in[7];
  const float* W_att2   = (const float*)d_in[8];
  const float* b_att2   = (const float*)d_in[9];

  float* A1 = (float*)d_ws;                       // 50000 x 64 f32
  float* A2 = A1 + (size_t)N_NODES * H2DIM;       // 50000 x 64 f32  (25.6MB total)

  const int nTiles = N_NODES / 16;                // 3125
  dim3 gridA((nTiles + WAVES - 1) / WAVES);
  node_transform_kernel<<<gridA, WAVES * 32, 0, stream>>>(
      features, W_nb, b_nb, W_self, b_self, W_att1, b_att1, A1, A2);

  const int nEdges = in_sizes[1] / 2;             // 600000
  edge_weight_kernel<<<(nEdges + 255) / 256, 256, 0, stream>>>(
      edges, A1, A2, W_att2, b_att2, (float*)d_out, nEdges);
}